// FCCapsuleLayer_2499670966845
// MI455X (gfx1250) — compile-verified
//
#include <hip/hip_runtime.h>
#include <math.h>

// ---------------------------------------------------------------------------
// FCCapsuleLayer on MI455X (gfx1250, wave32).
// Reference collapses (b never updated in routing loop) to:
//   s[b,j,c] = (1/64) * sum_{i,d} x[b,i,d] * W[j,i,c,d]   ; out = squash_c(s)
// GEMM: M=32 (batch), N=2048 (j*32+c), K=32768 (i*16+d).
// Memory bound: W = 256 MiB streamed once -> ~11.5 us floor @ 23.3 TB/s.
// Use V_WMMA_F32_16X16X4_F32 (fp32 inputs; converting the streamed operand
// would cost VALU without saving bytes). Split-K=64 + f32 atomics into d_ws.
// ---------------------------------------------------------------------------

typedef float v2f __attribute__((ext_vector_type(2)));
typedef float v8f __attribute__((ext_vector_type(8)));

#define NUM_CAPS 64
#define NUM_IN   2048
#define IN_DIM   16
#define CAPS_DIM 32
#define BATCH    32
#define KDIM     (NUM_IN * IN_DIM)        // 32768
#define NCOLS    (NUM_CAPS * CAPS_DIM)    // 2048
#define SEGS     64
#define KSEG     (KDIM / SEGS)            // 512 K per segment
#define KSTEPS   (KSEG / 4)               // 128 WMMA steps of K=4

// ---- pass 0: zero the fp32 accumulator (ws is poisoned by harness) --------
__global__ void caps_zero(float* __restrict__ ws) {
  ws[blockIdx.x * blockDim.x + threadIdx.x] = 0.0f;
}

// ---- pass 1: split-K WMMA GEMM, atomic-accumulate into ws -----------------
// grid = (64 j-tiles, 8 seg-groups), block = 256 (8 waves; each wave = 1 seg)
__global__ __launch_bounds__(256) void caps_gemm(const float* __restrict__ x,
                                                 const float* __restrict__ W,
                                                 float* __restrict__ ws) {
  const int lane = threadIdx.x & 31;
  const int wave = threadIdx.x >> 5;
  const int j    = blockIdx.x;                 // 0..63  capsule (N-tile of 32)
  const int seg  = blockIdx.y * 8 + wave;      // 0..63  K segment
  const int m    = lane & 15;                  // A: M row / B: N col (16-wide)
  const int kh   = lane >> 4;                  // K-half selector

  // A (16x4 f32): lane holds rows m, VGPRs = K {2*kh, 2*kh+1} -> contiguous.
  const float* xp0 = x + (size_t)m * KDIM + 2 * kh;            // batch 0..15
  const float* xp1 = xp0 + (size_t)16 * KDIM;                  // batch 16..31
  // B (4x16 f32): lane holds col (c0+m), VGPRs = K {2*kh, 2*kh+1}; since
  // k = i*16+d with d fastest in W, the K pair is contiguous in memory.
  const float* wp = W + (size_t)j * (NUM_IN * CAPS_DIM * IN_DIM)
                      + (size_t)m * IN_DIM + 2 * kh;

  v8f c00 = {}, c01 = {}, c10 = {}, c11 = {};

  int k0 = seg * KSEG;
  for (int s = 0; s < KSTEPS; ++s, k0 += 4) {
    const int woff = ((k0 >> 4) << 9) + (k0 & 15);   // i*512 + d0 floats
    v2f a0 = *(const v2f*)(xp0 + k0);
    v2f a1 = *(const v2f*)(xp1 + k0);
    v2f b0 = *(const v2f*)(wp + woff);               // c-half 0
    v2f b1 = *(const v2f*)(wp + woff + 256);         // c-half 1 (+16 cols)
    __builtin_prefetch(wp + woff + 4096, 0, 1);      // 8 i-blocks (16KB) ahead

    c00 = __builtin_amdgcn_wmma_f32_16x16x4_f32(false, a0, false, b0,
                                                (short)0, c00, false, false);
    c01 = __builtin_amdgcn_wmma_f32_16x16x4_f32(false, a0, false, b1,
                                                (short)0, c01, false, false);
    c10 = __builtin_amdgcn_wmma_f32_16x16x4_f32(false, a1, false, b0,
                                                (short)0, c10, false, false);
    c11 = __builtin_amdgcn_wmma_f32_16x16x4_f32(false, a1, false, b1,
                                                (short)0, c11, false, false);
  }

  // C/D 16x16 f32 layout: VGPR v, lane L -> M = v + 8*(L>>4), N = L&15.
  const int  ncol = lane & 15;
  const int  mrow = 8 * kh;
  float* base = ws + (size_t)j * CAPS_DIM + ncol;    // ws[b*2048 + j*32 + c]
#pragma unroll
  for (int v = 0; v < 8; ++v) {
    const int b0r = mrow + v;
    float* p0 = base + (size_t)b0r * NCOLS;          // batch 0..15 rows
    float* p1 = base + (size_t)(b0r + 16) * NCOLS;   // batch 16..31 rows
    unsafeAtomicAdd(p0,      c00[v]);                // global_atomic_add_f32
    unsafeAtomicAdd(p0 + 16, c01[v]);
    unsafeAtomicAdd(p1,      c10[v]);
    unsafeAtomicAdd(p1 + 16, c11[v]);
  }
}

// ---- pass 2: s *= 1/64 (softmax of zeros over 64 caps), then squash -------
// One wave32 per (b,j) row; lane = c; square-sum via 32-lane shfl_xor tree.
__global__ __launch_bounds__(128) void caps_squash(const float* __restrict__ ws,
                                                   float* __restrict__ out) {
  const int lane = threadIdx.x & 31;
  const int wave = threadIdx.x >> 5;
  const int row  = blockIdx.x * 4 + wave;            // 0..2047 == b*64 + j
  float s = ws[(size_t)row * CAPS_DIM + lane] * (1.0f / 64.0f);
  float ssq = s * s;
#pragma unroll
  for (int msk = 16; msk >= 1; msk >>= 1) ssq += __shfl_xor(ssq, msk, 32);
  const float scale = ssq / ((1.0f + ssq) * sqrtf(ssq + 1e-7f));
  out[(size_t)row * CAPS_DIM + lane] = scale * s;
}

extern "C" void kernel_launch(void* const* d_in, const int* in_sizes, int n_in,
                              void* d_out, int out_size, void* d_ws, size_t ws_size,
                              hipStream_t stream) {
  const float* x  = (const float*)d_in[0];   // (32, 2048, 16) f32
  const float* W  = (const float*)d_in[1];   // (64, 2048, 32, 16) f32
  float* out      = (float*)d_out;           // (32, 64, 32) f32
  float* acc      = (float*)d_ws;            // 65536 f32 accumulator (256 KB)

  caps_zero  <<<(BATCH * NCOLS) / 256, 256, 0, stream>>>(acc);
  caps_gemm  <<<dim3(NUM_CAPS, SEGS / 8), 256, 0, stream>>>(x, W, acc);
  caps_squash<<<(BATCH * NUM_CAPS) / 4, 128, 0, stream>>>(acc, out);
}